// BiGramLanguageModel_21921513078880
// MI455X (gfx1250) — compile-verified
//
#include <hip/hip_runtime.h>
#include <cstdint>
#include <cstddef>

// out[bt, :] = W[:, x[bt]]  for bt in [0, 32768), V = 5000.
// Pure bandwidth problem: transpose W once (coalesced, NT reads), keep W^T hot
// in the 192MB L2, stream the 655MB output with NT stores via the CDNA5 async
// global<->LDS DMA path (ASYNCcnt).

#define GATHER_BLOCK 256

// ---------------------------------------------------------------------------
// Kernel 1: LDS-tiled transpose  Wt[c][r] = W[r][c]   (n x n, n = 5000)
// 32x32 tile, 32x8 threads, +1 padding column to kill LDS bank conflicts.
// Interior tiles take a uniform fast path (no per-element bounds checks).
// All offset arithmetic is 32-bit (n*n = 25M < 2^31).
// W reads are non-temporal (read-once) so they don't evict W^T from L2.
// ---------------------------------------------------------------------------
__global__ __launch_bounds__(256) void transpose_tile_kernel(
    const float* __restrict__ W, float* __restrict__ Wt, int n) {
  __shared__ float tile[32][33];

  const int tx = threadIdx.x;
  const int ty = threadIdx.y;
  const int x  = blockIdx.x * 32 + tx;   // column in W (coalesced read)
  const int y0 = blockIdx.y * 32 + ty;   // row in W
  const int xt  = blockIdx.y * 32 + tx;  // column in Wt (coalesced write)
  const int yt0 = blockIdx.x * 32 + ty;  // row in Wt

  const bool interior = (blockIdx.x * 32 + 32 <= n) && (blockIdx.y * 32 + 32 <= n);

  if (interior) {
    const unsigned base = (unsigned)y0 * (unsigned)n + (unsigned)x;
#pragma unroll
    for (int j = 0; j < 32; j += 8)
      tile[ty + j][tx] = __builtin_nontemporal_load(W + base + (unsigned)j * n);

    __syncthreads();

    const unsigned baset = (unsigned)yt0 * (unsigned)n + (unsigned)xt;
#pragma unroll
    for (int j = 0; j < 32; j += 8)
      Wt[baset + (unsigned)j * n] = tile[tx][ty + j];
  } else {
#pragma unroll
    for (int j = 0; j < 32; j += 8) {
      const int y = y0 + j;
      if (x < n && y < n)
        tile[ty + j][tx] =
            __builtin_nontemporal_load(W + (unsigned)y * (unsigned)n + (unsigned)x);
    }
    __syncthreads();
#pragma unroll
    for (int j = 0; j < 32; j += 8) {
      const int yt = yt0 + j;
      if (xt < n && yt < n)
        Wt[(unsigned)yt * (unsigned)n + (unsigned)xt] = tile[tx][ty + j];
    }
  }
}

// ---------------------------------------------------------------------------
// Kernel 2: row gather. One block per (b,t). Copies Wt[tok, 0:5000] -> out row
// via async DMA:  global -> LDS (b128)  ...s_wait_asynccnt 0...  LDS -> global
// (b128, non-temporal so the streaming output does not evict Wt from L2).
// 5000 floats = 1250 x 16B chunks; LDS row buffer = 20000 B (16 blocks/WGP).
// ---------------------------------------------------------------------------
__global__ __launch_bounds__(GATHER_BLOCK) void gather_rows_async_kernel(
    const int* __restrict__ x, const float* __restrict__ Wt,
    float* __restrict__ out, int V) {
  __shared__ float row[5000];  // 20000 bytes

  const int bt  = blockIdx.x;
  const int tok = x[bt];  // uniform -> scalar load

  const uint64_t src0 = (uint64_t)(uintptr_t)(Wt + (size_t)tok * V);
  const uint64_t dst0 = (uint64_t)(uintptr_t)(out + (size_t)bt * V);
  // Low 32 bits of the flat shared address == workgroup-relative LDS offset.
  const unsigned lds0 = (unsigned)(uintptr_t)(void*)row;

  const int n16 = V / 4;  // 1250 sixteen-byte chunks
  const int tid = threadIdx.x;

  // Stage the whole row into LDS with async DMA loads (ASYNCcnt-tracked).
#pragma unroll
  for (int c = 0; c < 5; ++c) {
    const int idx = c * GATHER_BLOCK + tid;
    if (idx < n16) {
      const unsigned la = lds0 + (unsigned)idx * 16u;
      const uint64_t ga = src0 + (uint64_t)idx * 16u;
      asm volatile("global_load_async_to_lds_b128 %0, %1, off"
                   :
                   : "v"(la), "v"(ga)
                   : "memory");
    }
  }

  // Each wave waits for its own async loads; its stores read only the LDS
  // bytes it wrote itself, so no cross-wave barrier is required.
  __builtin_amdgcn_s_wait_asynccnt(0);

  // Stream the row out with non-temporal async stores (keep Wt hot in L2).
#pragma unroll
  for (int c = 0; c < 5; ++c) {
    const int idx = c * GATHER_BLOCK + tid;
    if (idx < n16) {
      const unsigned la = lds0 + (unsigned)idx * 16u;
      const uint64_t ga = dst0 + (uint64_t)idx * 16u;
      asm volatile("global_store_async_from_lds_b128 %0, %1, off th:TH_STORE_NT"
                   :
                   : "v"(ga), "v"(la)
                   : "memory");
    }
  }
  // S_ENDPGM performs an implicit wait-idle, covering outstanding async stores.
}

// ---------------------------------------------------------------------------
// Fallback (only if workspace cannot hold W^T): direct strided column gather,
// coalesced NT writes, strided reads served out of L2 (W fits in 192MB L2).
// ---------------------------------------------------------------------------
__global__ __launch_bounds__(GATHER_BLOCK) void gather_direct_kernel(
    const int* __restrict__ x, const float* __restrict__ W,
    float* __restrict__ out, int V) {
  const int bt  = blockIdx.x;
  const int tok = x[bt];
  float* dst = out + (size_t)bt * V;
  for (int v = threadIdx.x; v < V; v += GATHER_BLOCK) {
    const float val = W[(size_t)v * V + tok];
    __builtin_nontemporal_store(val, dst + v);
  }
}

extern "C" void kernel_launch(void* const* d_in, const int* in_sizes, int n_in,
                              void* d_out, int out_size, void* d_ws, size_t ws_size,
                              hipStream_t stream) {
  const int*   x = (const int*)d_in[0];    // [B*T] token ids (int32)
  const float* W = (const float*)d_in[1];  // [V, V] float32
  float*     out = (float*)d_out;          // [B*T, V] float32

  const int V  = 5000;
  const int NT = in_sizes[0];  // B*T = 32768 rows

  const size_t wt_bytes = (size_t)V * (size_t)V * sizeof(float);  // 100 MB

  if (ws_size >= wt_bytes) {
    float* Wt = (float*)d_ws;
    dim3 tb(32, 8);
    dim3 tg((V + 31) / 32, (V + 31) / 32);
    transpose_tile_kernel<<<tg, tb, 0, stream>>>(W, Wt, V);
    gather_rows_async_kernel<<<NT, GATHER_BLOCK, 0, stream>>>(x, Wt, out, V);
  } else {
    gather_direct_kernel<<<NT, GATHER_BLOCK, 0, stream>>>(x, W, out, V);
  }
}